// receptive_attention_45715631898961
// MI455X (gfx1250) — compile-verified
//
#include <hip/hip_runtime.h>

typedef __attribute__((ext_vector_type(2))) float v2f;
typedef __attribute__((ext_vector_type(4))) float v4f;
typedef __attribute__((ext_vector_type(8))) float v8f;

#define L_SEQ     2048
#define DHEAD     64
#define RECEPTIVE 128
#define NT        (L_SEQ / 16)   // 128 tiles per dimension
#define BH        32             // B*H
#define WAVES_PER_BLOCK 8

// One wave32 computes one 16x16 output tile:
//   scores[bh, 16*ti + m, 16*tj + n] = sum_d Q[bh, i, d] * K[bh, j, d],  masked by |i-j| <= 128
// using chained V_WMMA_F32_16X16X4_F32 over D=64 (16 steps of K=4).
__global__ __launch_bounds__(256) void band_qk_wmma_kernel(const float* __restrict__ Q,
                                                           const float* __restrict__ K,
                                                           float* __restrict__ Out) {
    const int lane = threadIdx.x & 31;
    const int wave = threadIdx.x >> 5;

    const unsigned tile = blockIdx.x * WAVES_PER_BLOCK + wave;   // 0 .. 524287
    const int bh  = (int)(tile >> 14);          // / (128*128)
    const int rem = (int)(tile & 16383);
    const int ti  = rem >> 7;
    const int tj  = rem & (NT - 1);

    float* out_tile = Out + (((size_t)bh * L_SEQ + (size_t)ti * 16) * L_SEQ) + (size_t)tj * 16;

    const int dt  = ti - tj;
    const int adt = dt < 0 ? -dt : dt;

    if (adt > 8) {
        // Fully outside the band: zero-fill the 16x16 tile.
        // 32 lanes x 2 stores x 16B = 1 KB. Rows are 64B contiguous; lane covers
        // row (lane>>2) and (lane>>2)+8, cols (lane&3)*4 .. +3.
        v4f z = {0.0f, 0.0f, 0.0f, 0.0f};
        const int r0 = lane >> 2;          // 0..7
        const int c0 = (lane & 3) * 4;     // 0,4,8,12
        __builtin_nontemporal_store(z, (v4f*)(out_tile + (size_t)r0 * L_SEQ + c0));
        __builtin_nontemporal_store(z, (v4f*)(out_tile + (size_t)(r0 + 8) * L_SEQ + c0));
        return;
    }

    const float* q_base = Q + ((size_t)bh * L_SEQ + (size_t)ti * 16) * DHEAD;
    const float* k_base = K + ((size_t)bh * L_SEQ + (size_t)tj * 16) * DHEAD;

    // A (16x4 f32): lanes 0-15 -> M=lane, K=4k+{0,1}; lanes 16-31 -> M=lane-16, K=4k+{2,3}.
    // B (4x16 f32): same striping over K's rows (N instead of M). Both are float2 loads.
    const int row = lane & 15;
    const int hi2 = (lane >> 4) * 2;   // 0 or 2
    const float* qp = q_base + (size_t)row * DHEAD + hi2;
    const float* kp = k_base + (size_t)row * DHEAD + hi2;

    v8f acc = {};
#pragma unroll
    for (int k = 0; k < 16; ++k) {
        v2f a = *(const v2f*)(qp + 4 * k);
        v2f b = *(const v2f*)(kp + 4 * k);
        // 8 args: (neg_a, A, neg_b, B, c_mod, C, reuse_a, reuse_b)
        acc = __builtin_amdgcn_wmma_f32_16x16x4_f32(false, a, false, b,
                                                    (short)0, acc, false, false);
    }

    // C/D layout: VGPR r -> row m = r (lanes 0-15, n = lane) or m = r+8 (lanes 16-31, n = lane-16).
    // Apply band mask |i-j| <= RECEPTIVE branchlessly, then stream out.
    const int n     = lane & 15;
    const int mbase = (lane >> 4) * 8;
    const int dbase = 16 * dt + mbase - n;   // (i - j) for r = 0
#pragma unroll
    for (int r = 0; r < 8; ++r) {
        const int d = dbase + r;
        const float v = (d >= -RECEPTIVE && d <= RECEPTIVE) ? acc[r] : 0.0f;
        __builtin_nontemporal_store(v, out_tile + (size_t)(mbase + r) * L_SEQ + n);
    }
}

extern "C" void kernel_launch(void* const* d_in, const int* in_sizes, int n_in,
                              void* d_out, int out_size, void* d_ws, size_t ws_size,
                              hipStream_t stream) {
    const float* Q = (const float*)d_in[0];
    const float* K = (const float*)d_in[1];
    float* Out     = (float*)d_out;

    // Total tiles: BH * NT * NT = 32 * 128 * 128 = 524288; 8 waves (tiles) per 256-thread block.
    const unsigned total_tiles = (unsigned)BH * NT * NT;
    dim3 grid(total_tiles / WAVES_PER_BLOCK);
    dim3 block(256);
    hipLaunchKernelGGL(band_qk_wmma_kernel, grid, block, 0, stream, Q, K, Out);
}